// ChunkLevelFeatureEncoderNERCNN_14310831030947
// MI455X (gfx1250) — compile-verified
//
#include <hip/hip_runtime.h>

// ---------------------------------------------------------------------------
// ChunkLevelFeatureEncoderNERCNN for MI455X (gfx1250)
// Ragged gather -> conv1d(k=3,pad=1,768->768) + bias + ReLU -> scatter,
// as a tiled fp32 GEMM on V_WMMA_F32_16X16X4_F32 with double-buffered
// GLOBAL_LOAD_ASYNC_TO_LDS staging (ASYNCcnt pipeline).
// ---------------------------------------------------------------------------

typedef float v2f __attribute__((ext_vector_type(2)));
typedef float v8f __attribute__((ext_vector_type(8)));
typedef int   v4i_vs __attribute__((vector_size(16)));  // pointee type of async b128 builtin

constexpr int Bn = 32;    // batch
constexpr int Lx = 1024;  // max wordpiece len
constexpr int Dd = 768;   // hidden
constexpr int Cc = 256;   // max chunks
constexpr int Kk = 8;     // max chunk len

constexpr int TM   = 128; // M tile (rows = (chunk,k) slots)
constexpr int TN   = 64;  // N tile (output channels)
constexpr int TKc  = 32;  // K chunk (input channels per stage)
constexpr int PADW = 36;  // padded LDS row stride: 36*4B=144B (16B aligned, 4-bank skew)
constexpr int NS   = 3 * (Dd / TKc); // 72 pipeline stages (tap x d-chunk)

// ----------------------- async copy to LDS helpers --------------------------
__device__ __forceinline__ void async_ld_b128(const float* g, float* l) {
#if __has_builtin(__builtin_amdgcn_global_load_async_to_lds_b128)
  __builtin_amdgcn_global_load_async_to_lds_b128(
      (__attribute__((address_space(1))) v4i_vs*)(void*)(g),
      (__attribute__((address_space(3))) v4i_vs*)(void*)(l), 0, 0);
#else
  unsigned lo = (unsigned)(unsigned long long)
      (__attribute__((address_space(3))) void*)(l);
  asm volatile("global_load_async_to_lds_b128 %0, %1, off"
               :: "v"(lo), "v"(g) : "memory");
#endif
}

#if __has_builtin(__builtin_amdgcn_s_wait_asynccnt)
#define WAIT_ASYNC(n) __builtin_amdgcn_s_wait_asynccnt(n)
#else
#define WAIT_ASYNC(n) asm volatile("s_wait_asynccnt %0" :: "i"(n) : "memory")
#endif

// -------------------------- zero helper -------------------------------------
__global__ __launch_bounds__(256) void zero_f4(float4* __restrict__ p, int n4) {
  int i = blockIdx.x * 256 + threadIdx.x;
  if (i < n4) p[i] = make_float4(0.f, 0.f, 0.f, 0.f);
}

// ------------------- per-batch exclusive cumsum of chunk lens ---------------
__global__ __launch_bounds__(256) void scan_chunks(const int* __restrict__ lens,
                                                   int* __restrict__ starts,
                                                   int* __restrict__ ctrunc) {
  __shared__ int s[Cc];
  int b = blockIdx.x;
  int c = threadIdx.x;
  int v = lens[b * Cc + c];
  s[c] = v;
  __syncthreads();
  for (int off = 1; off < Cc; off <<= 1) {
    int t = (c >= off) ? s[c - off] : 0;
    __syncthreads();
    s[c] += t;
    __syncthreads();
  }
  starts[b * Cc + c] = s[c] - v;           // exclusive prefix
  ctrunc[b * Cc + c] = (v < Kk) ? v : Kk;  // min(len, K)
}

// ------------- weight transpose: (O,I,3) -> Wt[t][o][i] ---------------------
__global__ __launch_bounds__(256) void wprep(const float* __restrict__ w,
                                             float* __restrict__ wt) {
  int i = blockIdx.x * 256 + threadIdx.x;
  constexpr int NT = 3 * Dd * Dd;
  if (i < NT) {
    int t = i / (Dd * Dd);
    int r = i - t * (Dd * Dd);
    int o = r / Dd;
    int d = r - o * Dd;
    wt[i] = w[o * (Dd * 3) + d * 3 + t];
  }
}

// ----------------------------- WMMA GEMM ------------------------------------
// Block: 256 threads = 8 waves. Tile: 128(M) x 64(N).
// Wave w owns rows [w*16, w*16+16) and all 64 N columns (4 x 16x16 subtiles).
// 72 stages, double-buffered LDS, async DMA one stage ahead of compute.
__global__ __launch_bounds__(256)
void conv_gemm_wmma(const float* __restrict__ tok,
                    const int* __restrict__ starts,
                    const int* __restrict__ ctrunc,
                    const float* __restrict__ wt,      // [3][768][768] (t,o,i)
                    const float* __restrict__ bias,    // [768]
                    const float* __restrict__ zerobuf, // [768] zeros
                    float* __restrict__ out)           // [B][L][D]
{
  __shared__ __align__(16) float As[2][TM][PADW];  // A tiles: rows x in-chan
  __shared__ __align__(16) float Bs[2][TN][PADW];  // B tiles (N-major)
  __shared__ const float* rowPtr[3][TM];           // gathered row base per tap

  const int tid  = threadIdx.x;
  const int lane = tid & 31;
  const int wave = tid >> 5;
  const int m0 = blockIdx.x * TM;
  const int n0 = blockIdx.y * TN;
  const int r = lane & 15;   // frag row / frag col
  const int h = lane >> 4;   // K-pair select

  // per-thread DMA assignments
  const int arow  = tid >> 1;      // A: 2 threads per row
  const int ahalf = tid & 1;       //    each covers 16 of 32 in-channels
  const int bn    = tid >> 2;      // B: 4 threads per out-channel row
  const int bq    = tid & 3;       //    each covers 8 of 32 in-channels

  v8f acc[4];
#pragma unroll
  for (int i = 0; i < 4; ++i)
#pragma unroll
    for (int e = 0; e < 8; ++e) acc[i][e] = 0.f;

  // ---- gathered row pointers for all 3 conv taps (invalid -> zerobuf) ------
  if (tid < TM) {
    int m = m0 + tid;       // global (chunk,k) slot
    int g = m >> 3;         // global chunk id
    int k = m & 7;          // position in chunk
    int b = g >> 8;         // batch (C == 256)
    int tr = ctrunc[g];
    int st = starts[g];
#pragma unroll
    for (int t = 0; t < 3; ++t) {
      int kk = k + t - 1;   // conv tap input position
      const float* p = zerobuf;
      if (kk >= 0 && kk < tr) p = tok + (long long)(b * Lx + st + kk) * Dd;
      rowPtr[t][tid] = p;
    }
  }
  __syncthreads();

  // ---- async DMA issue for one stage ---------------------------------------
  auto issue = [&](int s2, int bf2) {
    int t2  = s2 / (Dd / TKc);
    int d02 = (s2 - t2 * (Dd / TKc)) * TKc;
    const float* abase = rowPtr[t2][arow] + d02;
#pragma unroll
    for (int i = 0; i < 4; ++i) {              // 4 x b128 = 16 floats
      int col = ahalf * 16 + i * 4;
      async_ld_b128(abase + col, &As[bf2][arow][col]);
    }
    const float* bbase = wt + (long long)(t2 * Dd + n0 + bn) * Dd + d02;
#pragma unroll
    for (int i = 0; i < 2; ++i) {              // 2 x b128 = 8 floats
      int col = bq * 8 + i * 4;
      async_ld_b128(bbase + col, &Bs[bf2][bn][col]);
    }
  };

  issue(0, 0);

  const int mr = wave * 16 + r;
  for (int s = 0; s < NS; ++s) {
    const int buf = s & 1;
    if (s + 1 < NS) {
      issue(s + 1, buf ^ 1);   // prefetch next stage into other buffer
      WAIT_ASYNC(6);           // wait for stage s's 6 DMAs (in-order completion)
    } else {
      WAIT_ASYNC(0);
    }
    __syncthreads();

    // 8 K-steps x 4 N-subtiles of V_WMMA_F32_16X16X4_F32
#pragma unroll
    for (int k0 = 0; k0 < TKc; k0 += 4) {
      // A 16x4 frag: lane(r,h) -> { A[r][k0+2h], A[r][k0+2h+1] }
      v2f a = *reinterpret_cast<const v2f*>(&As[buf][mr][k0 + 2 * h]);
#pragma unroll
      for (int ns2 = 0; ns2 < 4; ++ns2) {
        // B 4x16 frag (mirrored): lane(r,h) -> { B[k0+2h][n], B[k0+2h+1][n] }
        v2f bf = *reinterpret_cast<const v2f*>(&Bs[buf][ns2 * 16 + r][k0 + 2 * h]);
        acc[ns2] = __builtin_amdgcn_wmma_f32_16x16x4_f32(
            false, a, false, bf, (short)0, acc[ns2], false, false);
      }
    }
    __syncthreads();  // all waves done reading buf before it is re-filled
  }

  // ------------- epilogue: bias + ReLU + ragged scatter ---------------------
#pragma unroll
  for (int ns2 = 0; ns2 < 4; ++ns2) {
    int n = n0 + ns2 * 16 + r;      // output channel (C/D: N = lane&15)
    float bv = bias[n];
#pragma unroll
    for (int j = 0; j < 8; ++j) {   // C/D: VGPR j -> M = j + 8h
      int m = m0 + wave * 16 + 8 * h + j;
      int g = m >> 3;
      int k = m & 7;
      if (k < ctrunc[g]) {
        int b = g >> 8;
        long long row = (long long)(b * Lx + starts[g] + k);
        float v = acc[ns2][j] + bv;
        out[row * Dd + n] = v > 0.f ? v : 0.f;
      }
    }
  }
}

// ---------------------------------------------------------------------------
extern "C" void kernel_launch(void* const* d_in, const int* in_sizes, int n_in,
                              void* d_out, int out_size, void* d_ws, size_t ws_size,
                              hipStream_t stream) {
  const float* tok   = (const float*)d_in[0];  // [B,L,D] f32
  const int*   lens  = (const int*)d_in[1];    // [B,C] i32
  const float* convw = (const float*)d_in[2];  // [D,D,3] f32
  const float* convb = (const float*)d_in[3];  // [D] f32
  float* out = (float*)d_out;                  // [B,L,D] f32

  char* ws = (char*)d_ws;
  int*   starts  = (int*)ws;                       // 8192 ints (32 KB)
  int*   ctrunc  = (int*)(ws + Bn * Cc * 4);       // 8192 ints (32 KB)
  float* wt      = (float*)(ws + 65536);           // 3*768*768 floats (~6.8 MB)
  float* zerobuf = (float*)(ws + 65536 + (size_t)3 * Dd * Dd * 4); // 768 floats

  int n4 = (Bn * Lx * Dd) / 4;
  zero_f4<<<(n4 + 255) / 256, 256, 0, stream>>>((float4*)out, n4);
  zero_f4<<<1, 256, 0, stream>>>((float4*)zerobuf, Dd / 4);
  scan_chunks<<<Bn, Cc, 0, stream>>>(lens, starts, ctrunc);
  wprep<<<(3 * Dd * Dd + 255) / 256, 256, 0, stream>>>(convw, wt);

  dim3 grid((Bn * Cc * Kk) / TM, Dd / TN);         // 512 x 12
  conv_gemm_wmma<<<grid, 256, 0, stream>>>(tok, starts, ctrunc, wt, convb,
                                           zerobuf, out);
}